// Nav_64939905516231
// MI455X (gfx1250) — compile-verified
//
#include <hip/hip_runtime.h>

// VIN (value-iteration network) for MI455X / gfx1250, wave32.
//
// Algebra: r = conv5x5(m, w_eff) + b_eff  where w_eff = sum_c r_w[c]*encode_w[c]
//          q0 = qr = conv5x5(r, q_w);  step: q = qr + conv5x5(v, w), v = max_ch(q)
// Everything for one batch image lives in LDS (~203 KB < 320 KB/WGP); the 10
// recurrent steps run with only block-local barriers. Convs are implicit GEMMs
// on V_WMMA_F32_16X16X4_F32. The maze image is DMA'd into LDS by the Tensor
// Data Mover (tensor_load_to_lds + s_wait_tensorcnt), overlapped with setup.
// Channel-max reductions use DPP (quad_perm / row_half_mirror) instead of
// ds_bpermute to keep the sequential VI chain off the LDS pipe.

typedef __attribute__((ext_vector_type(2))) float        v2f;
typedef __attribute__((ext_vector_type(8))) float        v8f;
typedef __attribute__((ext_vector_type(4))) unsigned int u32x4;
typedef __attribute__((ext_vector_type(4))) int          i32x4;
typedef __attribute__((ext_vector_type(8))) int          i32x8;

#ifndef __has_builtin
#define __has_builtin(x) 0
#endif
#if __has_builtin(__builtin_amdgcn_tensor_load_to_lds) && \
    __has_builtin(__builtin_amdgcn_s_wait_tensorcnt)
#define HAVE_TDM 1
#else
#define HAVE_TDM 0
#endif

#define PADW 68          // 64 + 2*2 zero halo
#define NPIX 4096        // 64*64

// max across each 8-lane group, pure VALU via DPP (no LDS, no dscnt waits):
// xor1 = quad_perm[1,0,3,2], xor2 = quad_perm[2,3,0,1], xor{4-ish} = half mirror
__device__ __forceinline__ float dpp_max8(float x) {
    float t;
    t = __int_as_float(__builtin_amdgcn_update_dpp(
            0, __float_as_int(x), 0xB1 /*quad_perm [1,0,3,2]*/, 0xF, 0xF, true));
    x = fmaxf(x, t);
    t = __int_as_float(__builtin_amdgcn_update_dpp(
            0, __float_as_int(x), 0x4E /*quad_perm [2,3,0,1]*/, 0xF, 0xF, true));
    x = fmaxf(x, t);
    t = __int_as_float(__builtin_amdgcn_update_dpp(
            0, __float_as_int(x), 0x141 /*ROW_HALF_MIRROR*/, 0xF, 0xF, true));
    x = fmaxf(x, t);
    return x;
}

// ---- K0: collapse encode conv + 1x1 r conv into one 50-tap filter ----------
__global__ void vin_weff_kernel(const float* __restrict__ encode_w,
                                const float* __restrict__ encode_b,
                                const float* __restrict__ r_w,
                                float* __restrict__ weff /* 51 floats */)
{
    const int t = threadIdx.x;
    if (t < 50) {
        float acc = 0.f;
        for (int c = 0; c < 150; ++c)
            acc = fmaf(r_w[c], encode_w[c * 50 + t], acc);
        weff[t] = acc;
    } else if (t == 50) {
        float acc = 0.f;
        for (int c = 0; c < 150; ++c)
            acc = fmaf(r_w[c], encode_b[c], acc);
        weff[50] = acc;
    }
}

// ---- K1: per-batch LDS-resident value iteration with WMMA convs ------------
__global__ void __launch_bounds__(256)
vin_vi_kernel(const int*   __restrict__ maze,
              const float* __restrict__ emb,
              const float* __restrict__ weff,
              const float* __restrict__ q_w,     // [8,1,5,5]
              const float* __restrict__ w_w,     // [8,1,5,5]
              const float* __restrict__ fc_w,    // [4,8]
              float*       __restrict__ out)     // [B,64,64,4]
{
    extern __shared__ float smem[];
    float* rpad  = smem;                     // 68*68 padded r image
    float* vA    = rpad + PADW * PADW;       // 68*68 padded v (ping)
    float* vB    = vA   + PADW * PADW;       // 68*68 padded v (pong)
    float* qbuf  = vB   + PADW * PADW;       // [8][4096]: qr, then final q
    int*   mzlds = (int*)(qbuf + 8 * NPIX);  // 4096 i32 maze tile

    const int b    = blockIdx.x;
    const int tid  = threadIdx.x;
    const int lane = tid & 31;
    const int wv   = tid >> 5;               // wave id 0..7
    const int NN   = lane & 15;              // WMMA N column / A-row M
    const int hf   = lane >> 4;              // lane half -> K pair select

#if HAVE_TDM
    // ---- TDM: DMA maze[b] (64x64 i32 = 16KB) into LDS, async ------------
    if (wv == 0) {
        const unsigned long long ga =
            (unsigned long long)(uintptr_t)(maze + (size_t)b * NPIX);
        const unsigned int la = (unsigned int)(uintptr_t)mzlds; // LDS byte off
        u32x4 g0 = {0, 0, 0, 0};
        g0[0] = 1u;                                  // count=1 (valid D#)
        g0[1] = la;                                  // lds_addr      [63:32]
        g0[2] = (unsigned int)ga;                    // global_addr   [95:64]
        g0[3] = (unsigned int)(ga >> 32) | (2u << 30); // addr hi | type=2
        i32x8 g1 = {0, 0, 0, 0, 0, 0, 0, 0};
        g1[0] = 2 << 16;              // data_size=2 (4B), wg_mask=0
        g1[1] = (int)(4096u << 16);   // tensor_dim0[15:0]  -> bits 63:48
        g1[2] = 1 << 16;              // dim0 hi=0 | tensor_dim1=1
        g1[3] = (int)(4096u << 16);   // tile_dim0=4096     -> bits 127:112
        g1[4] = 1;                    // tile_dim1=1
        g1[5] = 4096;                 // tensor_dim0_stride lo
        i32x4 z4 = {0, 0, 0, 0};
#if defined(__clang_major__) && (__clang_major__ >= 23)
        i32x8 z8 = {0, 0, 0, 0, 0, 0, 0, 0};
        __builtin_amdgcn_tensor_load_to_lds(g0, g1, z4, z4, z8, 0);
#else
        __builtin_amdgcn_tensor_load_to_lds(g0, g1, z4, z4, 0);
#endif
    }
#endif

    // zero rpad + both v buffers (zero halo == SAME padding); overlaps DMA
    for (int i = tid; i < 3 * PADW * PADW; i += 256) smem[i] = 0.f;

    // ---- per-lane WMMA operand prep (ISA 16x16x4 f32 layout) --------------
    // A: lane l holds pixel M=l&15, taps k0=4t+2*(l>=16) (vgpr0), k0+1 (vgpr1)
    // B: lane l holds channel N=l&15, same K mapping. Taps 25..27 are zero.
    float bq0[7], bq1[7], bw0[7], bw1[7];
    int   ao0[7], ao1[7];
    bool  va0[7], va1[7];
#pragma unroll
    for (int t = 0; t < 7; ++t) {
        const int  k0  = 4 * t + 2 * hf, k1 = k0 + 1;
        const bool okn = NN < 8;
        bq0[t] = (okn && k0 < 25) ? q_w[NN * 25 + k0] : 0.f;
        bq1[t] = (okn && k1 < 25) ? q_w[NN * 25 + k1] : 0.f;
        bw0[t] = (okn && k0 < 25) ? w_w[NN * 25 + k0] : 0.f;
        bw1[t] = (okn && k1 < 25) ? w_w[NN * 25 + k1] : 0.f;
        va0[t] = k0 < 25;  va1[t] = k1 < 25;
        const int d0 = va0[t] ? k0 : 0, d1 = va1[t] ? k1 : 0;
        ao0[t] = (d0 / 5) * PADW + (d0 % 5);
        ao1[t] = (d1 / 5) * PADW + (d1 % 5);
    }

#if HAVE_TDM
    __builtin_amdgcn_s_wait_tensorcnt(0);    // wave0's DMA done; others trivial
#else
    for (int i = tid; i < NPIX; i += 256) mzlds[i] = maze[(size_t)b * NPIX + i];
#endif
    __syncthreads();

    // ---- r = conv5x5(embed(maze), w_eff) + b_eff --------------------------
    {
        const float bias = weff[50];
        for (int p = tid; p < NPIX; p += 256) {
            const int y = p >> 6, x = p & 63;
            float acc = bias;
            for (int dy = 0; dy < 5; ++dy) {
                const int yy = y + dy - 2;
                if (yy < 0 || yy > 63) continue;
                for (int dx = 0; dx < 5; ++dx) {
                    const int xx = x + dx - 2;
                    if (xx < 0 || xx > 63) continue;
                    const int m   = mzlds[yy * 64 + xx];
                    const int tap = dy * 5 + dx;
                    acc = fmaf(weff[tap],      emb[m * 2 + 0], acc);
                    acc = fmaf(weff[25 + tap], emb[m * 2 + 1], acc);
                }
            }
            rpad[(y + 2) * PADW + (x + 2)] = acc;
        }
    }
    __syncthreads();

    // ---- qr = conv5x5(r, q_w); v1 = max_ch(qr) ----------------------------
    for (int tile = wv; tile < 256; tile += 8) {     // 16-pixel row tiles
        const int y     = tile >> 2;
        const int xb    = (tile & 3) << 4;
        const int abase = y * PADW + xb + NN;        // padded A base address
        const int pbase = y * 64 + xb + 8 * hf;      // C/D pixel base
        v8f c = {};
#pragma unroll
        for (int t = 0; t < 7; ++t) {
            v2f a, bb;
            a[0]  = va0[t] ? rpad[abase + ao0[t]] : 0.f;
            a[1]  = va1[t] ? rpad[abase + ao1[t]] : 0.f;
            bb[0] = bq0[t]; bb[1] = bq1[t];
            c = __builtin_amdgcn_wmma_f32_16x16x4_f32(false, a, false, bb,
                                                      (short)0, c, false, false);
        }
        if (NN < 8) {
#pragma unroll
            for (int j = 0; j < 8; ++j) qbuf[NN * NPIX + pbase + j] = c[j];
        }
#pragma unroll
        for (int j = 0; j < 8; ++j) {                // v = max over 8 channels
            const float m = dpp_max8(c[j]);
            if (NN == 0) vA[(y + 2) * PADW + xb + 8 * hf + j + 2] = m;
        }
    }
    __syncthreads();

    // ---- 10 value-iteration steps: q = qr + conv5x5(v, w) -----------------
    const float* vs = vA;
    float*       vd = vB;
    for (int s = 1; s <= 10; ++s) {
        for (int tile = wv; tile < 256; tile += 8) {
            const int y     = tile >> 2;
            const int xb    = (tile & 3) << 4;
            const int abase = y * PADW + xb + NN;
            const int pbase = y * 64 + xb + 8 * hf;
            v8f c;
#pragma unroll
            for (int j = 0; j < 8; ++j)              // C := qr (0 in pad cols)
                c[j] = (NN < 8) ? qbuf[NN * NPIX + pbase + j] : 0.f;
#pragma unroll
            for (int t = 0; t < 7; ++t) {
                v2f a, bb;
                a[0]  = va0[t] ? vs[abase + ao0[t]] : 0.f;
                a[1]  = va1[t] ? vs[abase + ao1[t]] : 0.f;
                bb[0] = bw0[t]; bb[1] = bw1[t];
                c = __builtin_amdgcn_wmma_f32_16x16x4_f32(false, a, false, bb,
                                                          (short)0, c, false, false);
            }
            if (s < 10) {
#pragma unroll
                for (int j = 0; j < 8; ++j) {
                    const float m = dpp_max8(c[j]);
                    if (NN == 0) vd[(y + 2) * PADW + xb + 8 * hf + j + 2] = m;
                }
            } else if (NN < 8) {                     // last step: q -> qbuf
#pragma unroll
                for (int j = 0; j < 8; ++j) qbuf[NN * NPIX + pbase + j] = c[j];
            }
        }
        __syncthreads();
        const float* t2 = vs; vs = vd; vd = (float*)t2;
    }

    // ---- out[b,y,x,:] = q[b,y,x,:] @ fc_w^T -------------------------------
    float fw[32];
#pragma unroll
    for (int i = 0; i < 32; ++i) fw[i] = fc_w[i];
    float* outb = out + (size_t)b * NPIX * 4;
    for (int p = tid; p < NPIX; p += 256) {
        float q[8];
#pragma unroll
        for (int ch = 0; ch < 8; ++ch) q[ch] = qbuf[ch * NPIX + p];
#pragma unroll
        for (int a = 0; a < 4; ++a) {
            float acc = 0.f;
#pragma unroll
            for (int ch = 0; ch < 8; ++ch) acc = fmaf(q[ch], fw[a * 8 + ch], acc);
            outb[p * 4 + a] = acc;
        }
    }
}

extern "C" void kernel_launch(void* const* d_in, const int* in_sizes, int n_in,
                              void* d_out, int out_size, void* d_ws, size_t ws_size,
                              hipStream_t stream)
{
    const int*   maze     = (const int*)  d_in[0];   // [64,64,64] int32
    const float* emb      = (const float*)d_in[1];   // [3,2]
    const float* encode_w = (const float*)d_in[2];   // [150,2,5,5]
    const float* encode_b = (const float*)d_in[3];   // [150]
    const float* r_w      = (const float*)d_in[4];   // [1,150,1,1]
    const float* q_w      = (const float*)d_in[5];   // [8,1,5,5]
    const float* w_w      = (const float*)d_in[6];   // [8,1,5,5]
    const float* fc_w     = (const float*)d_in[7];   // [4,8]
    float* out  = (float*)d_out;                     // [64,64,64,4]
    float* weff = (float*)d_ws;                      // 51 floats scratch

    vin_weff_kernel<<<1, 64, 0, stream>>>(encode_w, encode_b, r_w, weff);

    const size_t smem_bytes =
        (size_t)(3 * PADW * PADW + 8 * NPIX) * sizeof(float) +
        (size_t)NPIX * sizeof(int);
    vin_vi_kernel<<<64, 256, smem_bytes, stream>>>(maze, emb, weff,
                                                   q_w, w_w, fc_w, out);
}